// Weighted_least_squares_80436147519858
// MI455X (gfx1250) — compile-verified
//
#include <hip/hip_runtime.h>

// Problem constants (from reference)
#define BS        16
#define NCLASSES  4
#define HW        (256 * 512)       // 131072
#define REG_LS    1e-3f

// Tiling
#define BPB       64                // blocks per batch
#define WAVES     8                 // waves per block
#define BLOCK_T   (WAVES * 32)      // 256 threads
#define NPB       (HW / BPB)        // 2048 pixels per block
#define NPW       (NPB / WAVES)     // 256 pixels per wave
#define PXI       128               // pixels per wave-iteration (4 per lane)
#define NITER     (NPW / PXI)       // 2
#define STRIDE    136               // LDS row stride: banks = 8*row + col (mod 64)

typedef __attribute__((ext_vector_type(2))) float v2f;
typedef __attribute__((ext_vector_type(8))) float v8f;

// Per (batch, class): S[c][m] = sum_n Wsq[c][n] * mom[m][n] as a 4xK * Kx8
// GEMM on v_wmma_f32_16x16x4_f32 (full fp32). Streaming is b128 end to end.
__global__ __launch_bounds__(BLOCK_T) void wls_moments_kernel(
    const float* __restrict__ Wp,     // (BS, NCLASSES, HW)
    const float* __restrict__ grid,   // (BS, HW, 2)
    float* __restrict__ ws)           // (BS, BPB, 32) block partials
{
    __shared__ __align__(16) float wsqA[WAVES][NCLASSES][STRIDE]; // A rows
    __shared__ __align__(16) float momB[WAVES][8][STRIDE];        // B rows
    __shared__ __align__(16) float zbuf[STRIDE];                  // zero row
    __shared__ float red[WAVES][32];

    const int blk  = blockIdx.x;            // 0..BPB-1
    const int b    = blockIdx.y;            // 0..BS-1
    const int tid  = threadIdx.x;
    const int wv   = tid >> 5;
    const int lane = tid & 31;
    const int row  = lane & 15;             // M (A) / N (B) index
    const int csel = (lane >= 16) ? 4 : 0;  // hi half-wave -> upper K columns
    const int nl   = lane * 4;              // this lane's 4 staging columns

    // One-time init: zero row for dead lanes, constant-ones moment row (m=4).
    for (int i = tid; i < STRIDE; i += BLOCK_T) zbuf[i] = 0.0f;
    for (int i = tid; i < WAVES * STRIDE; i += BLOCK_T)
        momB[i / STRIDE][4][i % STRIDE] = 1.0f;
    __syncthreads();

    const float* __restrict__ Wb = Wp + (size_t)b * NCLASSES * HW;
    const float* __restrict__ Gb = grid + (size_t)b * HW * 2;
    const int n_base = blk * NPB + wv * NPW;

    // Divergence-free fragment base pointers: dead lanes broadcast-read zbuf.
    const float* aPtr = (row < NCLASSES) ? &wsqA[wv][row][0] : &zbuf[0];
    const float* bPtr = (row < 8)        ? &momB[wv][row][0] : &zbuf[0];

    v8f acc = {};  // live region: VGPRs 0..3 (class), lanes 0..7 (moment)

    for (int it = 0; it < NITER; ++it) {
        const int n0 = n_base + it * PXI;

        // ---- stream 4 pixels/lane with b128 loads ----
        const float* wp = Wb + n0 + nl;
        const float4 w0 = *(const float4*)(wp);
        const float4 w1 = *(const float4*)(wp + HW);
        const float4 w2 = *(const float4*)(wp + 2 * HW);
        const float4 w3 = *(const float4*)(wp + 3 * HW);
        const float* gp = Gb + (size_t)(n0 + nl) * 2;
        const float4 gA = *(const float4*)(gp);        // x0 y0 x1 y1
        const float4 gB = *(const float4*)(gp + 4);    // x2 y2 x3 y3

        const float X[4]  = {gA.x, gA.z, gB.x, gB.z};
        const float Y[4]  = {255.0f - gA.y, 255.0f - gA.w,
                             255.0f - gB.y, 255.0f - gB.w};
        float Y2[4], Y3[4], Y4[4], Y2X[4], YX[4];
        #pragma unroll
        for (int j = 0; j < 4; ++j) {
            Y2[j]  = Y[j] * Y[j];
            Y3[j]  = Y2[j] * Y[j];
            Y4[j]  = Y2[j] * Y2[j];
            Y2X[j] = Y2[j] * X[j];
            YX[j]  = Y[j] * X[j];
        }

        // ---- stage transposed, b128 stores ----
        *(float4*)&wsqA[wv][0][nl] = make_float4(w0.x*w0.x, w0.y*w0.y, w0.z*w0.z, w0.w*w0.w);
        *(float4*)&wsqA[wv][1][nl] = make_float4(w1.x*w1.x, w1.y*w1.y, w1.z*w1.z, w1.w*w1.w);
        *(float4*)&wsqA[wv][2][nl] = make_float4(w2.x*w2.x, w2.y*w2.y, w2.z*w2.z, w2.w*w2.w);
        *(float4*)&wsqA[wv][3][nl] = make_float4(w3.x*w3.x, w3.y*w3.y, w3.z*w3.z, w3.w*w3.w);
        *(float4*)&momB[wv][0][nl] = make_float4(Y4[0], Y4[1], Y4[2], Y4[3]);
        *(float4*)&momB[wv][1][nl] = make_float4(Y3[0], Y3[1], Y3[2], Y3[3]);
        *(float4*)&momB[wv][2][nl] = make_float4(Y2[0], Y2[1], Y2[2], Y2[3]);
        *(float4*)&momB[wv][3][nl] = make_float4(Y[0], Y[1], Y[2], Y[3]);
        *(float4*)&momB[wv][5][nl] = make_float4(Y2X[0], Y2X[1], Y2X[2], Y2X[3]);
        *(float4*)&momB[wv][6][nl] = make_float4(YX[0], YX[1], YX[2], YX[3]);
        *(float4*)&momB[wv][7][nl] = make_float4(X[0], X[1], X[2], X[3]);

        __syncthreads();

        // ---- WMMA phase: one b128 LDS load per fragment feeds 2 wmma steps.
        // Step pair s: lo lanes own pixels (cols) 8s..8s+3, hi lanes 8s+4..8s+7.
        #pragma unroll
        for (int s = 0; s < PXI / 8; ++s) {
            const int col = 8 * s + csel;
            const float4 av = *(const float4*)(aPtr + col);
            const float4 bv = *(const float4*)(bPtr + col);
            v2f a0 = {av.x, av.y}, b0 = {bv.x, bv.y};
            acc = __builtin_amdgcn_wmma_f32_16x16x4_f32(
                false, a0, false, b0, (short)0, acc, false, false);
            v2f a1 = {av.z, av.w}, b1 = {bv.z, bv.w};
            acc = __builtin_amdgcn_wmma_f32_16x16x4_f32(
                false, a1, false, b1, (short)0, acc, false, false);
        }

        __syncthreads();
    }

    // ---- deterministic in-block reduction of the 4x8 live corner ----
    // C/D layout: VGPR r, lanes 0..15 => M=r (class), N=lane (moment).
    if (lane < 8) {
        red[wv][0 * 8 + lane] = acc[0];
        red[wv][1 * 8 + lane] = acc[1];
        red[wv][2 * 8 + lane] = acc[2];
        red[wv][3 * 8 + lane] = acc[3];
    }
    __syncthreads();

    if (tid < 32) {   // tid = c*8 + m
        float s = 0.0f;
        #pragma unroll
        for (int w = 0; w < WAVES; ++w) s += red[w][tid];
        ws[((size_t)b * BPB + blk) * 32 + tid] = s;
    }
}

// Final: sum block partials in fixed order, build Z + reg*I, solve 3x3 by
// adjugate, scatter betas in (class, batch, coeff) output order.
__global__ void wls_solve_kernel(const float* __restrict__ ws,
                                 float* __restrict__ out)
{
    const int t = threadIdx.x;
    if (t >= BS * NCLASSES) return;
    const int b = t >> 2;
    const int c = t & 3;

    float S[8] = {0.f, 0.f, 0.f, 0.f, 0.f, 0.f, 0.f, 0.f};
    for (int blk = 0; blk < BPB; ++blk) {
        const float* p = ws + ((size_t)b * BPB + blk) * 32 + c * 8;
        #pragma unroll
        for (int m = 0; m < 8; ++m) S[m] += p[m];
    }

    // S0=w2y4 S1=w2y3 S2=w2y2 S3=w2y S4=w2 S5=w2y2x S6=w2yx S7=w2x
    const float z00 = S[0] + REG_LS;
    const float z01 = S[1];
    const float z02 = S[2];
    const float z11 = S[2] + REG_LS;
    const float z12 = S[3];
    const float z22 = S[4] + REG_LS;
    const float X0 = S[5], X1 = S[6], X2 = S[7];

    const float A00 = z11 * z22 - z12 * z12;
    const float A01 = z02 * z12 - z01 * z22;
    const float A02 = z01 * z12 - z02 * z11;
    const float A11 = z00 * z22 - z02 * z02;
    const float A12 = z02 * z01 - z00 * z12;
    const float A22 = z00 * z11 - z01 * z01;
    const float det = z00 * A00 + z01 * A01 + z02 * A02;
    const float inv = 1.0f / det;

    const float b0 = (A00 * X0 + A01 * X1 + A02 * X2) * inv;
    const float b1 = (A01 * X0 + A11 * X1 + A12 * X2) * inv;
    const float b2 = (A02 * X0 + A12 * X1 + A22 * X2) * inv;

    float* o = out + (size_t)c * (BS * 3) + (size_t)b * 3;
    o[0] = b0;
    o[1] = b1;
    o[2] = b2;
}

extern "C" void kernel_launch(void* const* d_in, const int* in_sizes, int n_in,
                              void* d_out, int out_size, void* d_ws, size_t ws_size,
                              hipStream_t stream) {
    const float* W    = (const float*)d_in[0];   // (16, 4, 131072) f32
    const float* grid = (const float*)d_in[1];   // (16, 131072, 2) f32
    float* out = (float*)d_out;                  // 192 f32
    float* ws  = (float*)d_ws;                   // 16*64*32 f32 = 128 KB

    dim3 g(BPB, BS);
    wls_moments_kernel<<<g, BLOCK_T, 0, stream>>>(W, grid, ws);
    wls_solve_kernel<<<1, 64, 0, stream>>>(ws, out);
}